// Quanvolution__gen301_65481071395943
// MI455X (gfx1250) — compile-verified
//
#include <hip/hip_runtime.h>
#include <hip/hip_bf16.h>

typedef float v2f __attribute__((ext_vector_type(2)));
typedef float v8f __attribute__((ext_vector_type(8)));

#define WIRES 4
#define NPATCH 196
#define BATCH 8192
#define NTILES ((BATCH * NPATCH) / 16)   // 100352 tiles of 16 patches

// ---------------------------------------------------------------------------
// Kernel 1: build the fixed 16x16 circuit matrix (post-encoding part) from
// params. One lane per column: apply circuit to basis vector e_j.
// State index i = b0*8 + b1*4 + b2*2 + b3 (wire 0 = MSB), stride_w = 8>>w.
// ---------------------------------------------------------------------------
__global__ void quanv_build_matrix(const float* __restrict__ params,
                                   float* __restrict__ Mws) {
    const int j = threadIdx.x;
    if (j >= 16) return;

    float v[16];
#pragma unroll
    for (int i = 0; i < 16; ++i) v[i] = (i == j) ? 1.0f : 0.0f;

    auto cnot = [&](int ctrl, int tgt) {
        const int sc = 8 >> ctrl, st = 8 >> tgt;
#pragma unroll
        for (int i = 0; i < 16; ++i)
            if ((i & sc) && !(i & st)) {
                float tmp = v[i]; v[i] = v[i | st]; v[i | st] = tmp;
            }
    };
    auto ry = [&](float th, int w) {
        const int st = 8 >> w;
        const float cth = __cosf(0.5f * th);
        const float sth = __sinf(0.5f * th);
#pragma unroll
        for (int i = 0; i < 16; ++i)
            if (!(i & st)) {
                float a = v[i], b = v[i | st];
                v[i]      = cth * a - sth * b;
                v[i | st] = sth * a + cth * b;
            }
    };

    // entangler after data encoding
    cnot(0, 1); cnot(1, 2); cnot(2, 3);
    // variational layers
#pragma unroll
    for (int layer = 0; layer < 2; ++layer) {
#pragma unroll
        for (int w = 0; w < WIRES; ++w) ry(params[layer * WIRES + w], w);
        cnot(0, 1); cnot(1, 2); cnot(2, 3);
    }

#pragma unroll
    for (int i = 0; i < 16; ++i) Mws[i * 16 + j] = v[i];   // row-major M[i][j]
}

// ---------------------------------------------------------------------------
// Kernel 2: streaming quanvolution. Each wave handles tiles of 16 patches:
//   B (4x16 per K-chunk) = kron-encoded states of 16 patches (columns)
//   A (16x4 per K-chunk) = Mcirc column chunk
//   D = sum_kc A_kc x B_kc  (4x V_WMMA_F32_16X16X4_F32)
// D layout: VGPR r, lanes 0-15 -> (row r,   patch=lane)
//                   lanes16-31 -> (row r+8, patch=lane-16)
//
// B-fragment element for this lane: s0[k], k = kc*4 + half*2 + j.
// Since s0[k] = t01[k>>2] * t23[k&3], and (k&3) = half*2 + j:
//   t23[half*2 + j] = (half ? sin2 : cos2) * (j ? sin3 : cos3)
// -> one cndmask + pure multiplies, no dynamic array indexing.
// ---------------------------------------------------------------------------
__global__ void quanv_main(const float* __restrict__ x,
                           const float* __restrict__ Mws,
                           float* __restrict__ out) {
    const int lane = threadIdx.x & 31;
    const int col  = lane & 15;   // patch slot / matrix row (M) for A
    const int half = lane >> 4;   // selects K sub-pair per ISA f32 A/B layout

    // Load A fragments once per wave: A[kc] holds Mcirc[col][k], k = kc*4 + half*2 + j
    v2f A[4];
#pragma unroll
    for (int kc = 0; kc < 4; ++kc) {
#pragma unroll
        for (int jj = 0; jj < 2; ++jj) {
            const int k = kc * 4 + half * 2 + jj;
            A[kc][jj] = Mws[col * 16 + k];
        }
    }

    const int wid = blockIdx.x * (blockDim.x >> 5) + (threadIdx.x >> 5);
    const int nw  = gridDim.x * (blockDim.x >> 5);

    for (int tile = wid; tile < NTILES; tile += nw) {
        const int pg = tile * 16 + col;          // global patch id
        const int b  = pg / NPATCH;
        const int p  = pg - b * NPATCH;
        const int pr = p / 14;
        const int pc = p - pr * 14;

        // 2x2 patch pixels: rows 2*pr, 2*pr+1; cols 2*pc, 2*pc+1 (28x28 image)
        const float* px = x + (size_t)b * 784 + pr * 56 + pc * 2;
        const float2 top = *(const float2*)(px);
        const float2 bot = *(const float2*)(px + 28);

        float cw0, sw0, cw1, sw1, cw2, sw2, cw3, sw3;
        {
            float a0 = 0.5f * top.x, a1 = 0.5f * top.y;
            float a2 = 0.5f * bot.x, a3 = 0.5f * bot.y;
            sw0 = __sinf(a0); cw0 = __cosf(a0);
            sw1 = __sinf(a1); cw1 = __cosf(a1);
            sw2 = __sinf(a2); cw2 = __cosf(a2);
            sw3 = __sinf(a3); cw3 = __cosf(a3);
        }

        // wires 0,1 kron factors (compile-time indexed by kc)
        const float t01[4] = { cw0 * cw1, cw0 * sw1, sw0 * cw1, sw0 * sw1 };
        // wires 2,3 kron factors for this lane's half: one select, two mults
        const float w2 = half ? sw2 : cw2;
        const float ta = w2 * cw3;   // t23[half*2 + 0]
        const float tb = w2 * sw3;   // t23[half*2 + 1]

        v2f B[4];
#pragma unroll
        for (int kc = 0; kc < 4; ++kc) {
            B[kc][0] = t01[kc] * ta;
            B[kc][1] = t01[kc] * tb;
        }

        // v = Mcirc * s0 for 16 patches via 4 chained f32 WMMAs
        v8f acc = {};
#pragma unroll
        for (int kc = 0; kc < 4; ++kc) {
            acc = __builtin_amdgcn_wmma_f32_16x16x4_f32(
                false, A[kc], false, B[kc], (short)0, acc, false, false);
        }

        // <Z_w> = T - 2*U_w ; this lane owns rows i = half*8 + r of its patch.
        // Wire-0 bit of i is just `half` -> U0 = half ? T_partial : 0.
        float T = 0.f, U1 = 0.f, U2 = 0.f, U3 = 0.f;
#pragma unroll
        for (int r = 0; r < 8; ++r) {
            const float pv = acc[r] * acc[r];
            T += pv;
            if (r & 4) U1 += pv;
            if (r & 2) U2 += pv;
            if (r & 1) U3 += pv;
        }
        float U0 = half ? T : 0.0f;

        T  += __shfl_xor(T, 16, 32);
        U0 += __shfl_xor(U0, 16, 32);
        U1 += __shfl_xor(U1, 16, 32);
        U2 += __shfl_xor(U2, 16, 32);
        U3 += __shfl_xor(U3, 16, 32);

        if (half == 0) {
            float4 z = { T - 2.f * U0, T - 2.f * U1, T - 2.f * U2, T - 2.f * U3 };
            *(float4*)(out + (size_t)b * (NPATCH * WIRES) + p * 4) = z;
        }
    }
}

extern "C" void kernel_launch(void* const* d_in, const int* in_sizes, int n_in,
                              void* d_out, int out_size, void* d_ws, size_t ws_size,
                              hipStream_t stream) {
    const float* x      = (const float*)d_in[0];   // (8192,1,28,28) f32
    const float* params = (const float*)d_in[1];   // (2,4) f32
    float* out = (float*)d_out;                    // (8192, 784) f32
    float* Mws = (float*)d_ws;                     // 16x16 circuit matrix

    quanv_build_matrix<<<1, 32, 0, stream>>>(params, Mws);

    const int threads = 256;                       // 8 waves per block
    const int blocks  = 1568;                      // ~12.5k waves, grid-stride
    quanv_main<<<blocks, threads, 0, stream>>>(x, Mws, out);
}